// SinkhornOptimizer_v2_45827301048810
// MI455X (gfx1250) — compile-verified
//
#include <hip/hip_runtime.h>

// Sinkhorn-Knopp via dual potentials:
//   u[i] = lse_j(X[i,j]*S - v[j]);  v[j] = lse_i(X[i,j]*S - u[i]);  P = exp(X*S - u - v)
// X (128MB) is read-only and L2-resident across all 200 passes on MI455X (192MB L2).

#define SK_B 128
#define SK_N 512
#define SK_SCALE 10.0f   // 1 / ENTROPY_REG
#define SK_ITERS 100

typedef __attribute__((ext_vector_type(2))) float v2f;
typedef __attribute__((ext_vector_type(8))) float v8f;

__global__ void sk_zero(float* __restrict__ p, int count) {
  int i = blockIdx.x * blockDim.x + threadIdx.x;
  if (i < count) p[i] = 0.0f;
}

// Row pass: each wave owns 16 rows. Pass 1: per-row max (VALU + shfl reduce).
// Pass 2: row-sum of exp via V_WMMA_F32_16X16X4_F32 with B = ones (matrix pipe
// does the accumulation, co-executing with v_exp_f32 on the VALU).
__global__ __launch_bounds__(256) void sk_row(const float* __restrict__ X,
                                              const float* __restrict__ v,
                                              float* __restrict__ u) {
  __shared__ float vs[SK_N];
  const int n    = SK_N;
  const int bidx = blockIdx.x >> 2;   // matrix index
  const int rblk = blockIdx.x & 3;    // 128-row block
  const int tid  = threadIdx.x;

  vs[tid]       = v[bidx * n + tid];
  vs[tid + 256] = v[bidx * n + tid + 256];
  __syncthreads();

  const int wave = tid >> 5;
  const int lane = tid & 31;
  const int r0   = rblk * 128 + wave * 16;      // first of this wave's 16 rows
  const float* Xb = X + (size_t)bidx * n * n;

  // ---- pass 1: m_i = max_j (X[i,j]*S - v[j]) for the 16 rows ----
  const int rowA = lane & 15;                    // row this lane feeds into WMMA A
  const int rowW = (lane < 16) ? lane : (lane - 8); // row this lane writes (lanes 0-7,16-23)
  float m_a = -__builtin_inff();
  float m_w = -__builtin_inff();
#pragma unroll
  for (int i = 0; i < 16; ++i) {
    const float* row = Xb + (size_t)(r0 + i) * n;
    float m = -__builtin_inff();
#pragma unroll
    for (int j = lane; j < n; j += 32)
      m = fmaxf(m, row[j] * SK_SCALE - vs[j]);
#pragma unroll
    for (int off = 16; off > 0; off >>= 1)
      m = fmaxf(m, __shfl_xor(m, off, 32));
    m_a = (rowA == i) ? m : m_a;
    m_w = (rowW == i) ? m : m_w;
  }

  // ---- pass 2: acc[m, :] = sum_j exp(X[m,j]*S - v[j] - m_m) via WMMA ----
  // 32-bit A 16x4 layout: lanes 0-15 hold M=lane, K=0,1; lanes 16-31 hold K=2,3.
  v8f acc = {};
  v2f bones; bones.x = 1.0f; bones.y = 1.0f;     // B = ones -> D = row sums of A
  const int koff = (lane >> 4) * 2;
  const float* arow = Xb + (size_t)(r0 + rowA) * n + koff;
  for (int j = 0; j < n; j += 4) {
    v2f x = *(const v2f*)(arow + j);             // 8B aligned (koff even)
    v2f a;
    a.x = __expf(x.x * SK_SCALE - vs[j + koff]     - m_a);
    a.y = __expf(x.y * SK_SCALE - vs[j + koff + 1] - m_a);
    acc = __builtin_amdgcn_wmma_f32_16x16x4_f32(false, a, false, bones,
                                                (short)0, acc, false, false);
  }

  // ---- write u: D layout -> VGPR r holds M=r (lanes 0-15) and M=8+r (lanes 16-31)
  float* ub = u + bidx * n + r0;
#pragma unroll
  for (int i = 0; i < 8; ++i) {
    float s = acc[i];
    if (lane == i)      ub[i]     = m_w + __logf(s);
    if (lane == 16 + i) ub[8 + i] = m_w + __logf(s);
  }
}

// Col pass: one thread per column, two-pass lse streaming down the rows
// (coalesced across lanes, L2-resident), u staged in LDS.
__global__ __launch_bounds__(256) void sk_col(const float* __restrict__ X,
                                              const float* __restrict__ u,
                                              float* __restrict__ v) {
  __shared__ float us[SK_N];
  const int n    = SK_N;
  const int bidx = blockIdx.x >> 1;
  const int cblk = blockIdx.x & 1;
  const int tid  = threadIdx.x;

  us[tid]       = u[bidx * n + tid];
  us[tid + 256] = u[bidx * n + tid + 256];
  __syncthreads();

  const int col = cblk * 256 + tid;
  const float* Xc = X + (size_t)bidx * n * n + col;

  float m = -__builtin_inff();
#pragma unroll 8
  for (int i = 0; i < n; ++i) {
    if (i + 16 < n) __builtin_prefetch(Xc + (size_t)(i + 16) * n, 0, 3);
    m = fmaxf(m, Xc[(size_t)i * n] * SK_SCALE - us[i]);
  }
  float s = 0.0f;
#pragma unroll 8
  for (int i = 0; i < n; ++i)
    s += __expf(Xc[(size_t)i * n] * SK_SCALE - us[i] - m);

  v[bidx * n + col] = m + __logf(s);
}

// Final: P = exp(X*S - u_i - v_j), streamed coalesced to d_out.
__global__ __launch_bounds__(256) void sk_final(const float* __restrict__ X,
                                                const float* __restrict__ u,
                                                const float* __restrict__ v,
                                                float* __restrict__ P) {
  const int n    = SK_N;
  const int row  = blockIdx.x;
  const int bidx = blockIdx.y;
  const size_t base = ((size_t)bidx * n + row) * n;
  const float ui = u[bidx * n + row];
  const int j0 = threadIdx.x;

  float x0 = X[base + j0];
  float x1 = X[base + j0 + 256];
  float w0 = v[bidx * n + j0];
  float w1 = v[bidx * n + j0 + 256];
  P[base + j0]       = __expf(x0 * SK_SCALE - ui - w0);
  P[base + j0 + 256] = __expf(x1 * SK_SCALE - ui - w1);
}

extern "C" void kernel_launch(void* const* d_in, const int* in_sizes, int n_in,
                              void* d_out, int out_size, void* d_ws, size_t ws_size,
                              hipStream_t stream) {
  const float* X = (const float*)d_in[0];
  float* out = (float*)d_out;
  const int B = SK_B, n = SK_N;

  float* u  = (float*)d_ws;             // B*n floats
  float* vv = u + (size_t)B * n;        // B*n floats  (512 KB total in d_ws)

  sk_zero<<<(B * n + 255) / 256, 256, 0, stream>>>(vv, B * n);

  for (int t = 0; t < SK_ITERS; ++t) {
    sk_row<<<B * 4, 256, 0, stream>>>(X, vv, u);   // u = lse_rows(X*S - v)
    sk_col<<<B * 2, 256, 0, stream>>>(X, u, vv);   // v = lse_cols(X*S - u)
  }

  dim3 grid(n, B);
  sk_final<<<grid, 256, 0, stream>>>(X, u, vv, out);
}